// Squeeze_Exicitation_59717225284138
// MI455X (gfx1250) — compile-verified
//
#include <hip/hip_runtime.h>
#include <math.h>

// Squeeze-Excitation fused kernel for MI455X (gfx1250, wave32).
// B=8192, C=2048, H=W=2, CR=128.

typedef float v2f __attribute__((ext_vector_type(2)));
typedef float v8f __attribute__((ext_vector_type(8)));

#define BATCH   8192
#define CH      2048
#define CRD     128
#define ROWS    16          // batch rows per block
#define PSTR    2052        // padded LDS row stride for p/g (breaks bank conflicts)
#define NWAVES  8
#define NTHREAD 256
#define SE_EPS  1e-5f

// LDS layout (floats):
//   [0,            CH)              sh_inv
//   [CH,           2*CH)            sh_shift
//   [2*CH,         2*CH+ROWS*PSTR)  p (later reused as g)
//   [.. ,          + ROWS*CRD)      h
#define SMEM_FLOATS (2*CH + ROWS*PSTR + ROWS*CRD)

__global__ __launch_bounds__(NTHREAD)
void se_fused_kernel(const float* __restrict__ x,
                     const float* __restrict__ gamma,
                     const float* __restrict__ beta,
                     const float* __restrict__ mean,
                     const float* __restrict__ var,
                     const float* __restrict__ w1,
                     const float* __restrict__ b1,
                     const float* __restrict__ w2,
                     const float* __restrict__ b2,
                     float* __restrict__ out)
{
    extern __shared__ float smem[];
    float* sh_inv   = smem;
    float* sh_shift = smem + CH;
    float* p_lds    = smem + 2 * CH;                 // [ROWS][PSTR], reused for g
    float* h_lds    = smem + 2 * CH + ROWS * PSTR;   // [ROWS][CRD]

    const int tid = threadIdx.x;
    const int b0  = blockIdx.x * ROWS;

    // ---- Phase 0: per-channel BN coefficients into LDS --------------------
    for (int c = tid; c < CH; c += NTHREAD) {
        float inv = gamma[c] * rsqrtf(var[c] + SE_EPS);
        sh_inv[c]   = inv;
        sh_shift[c] = beta[c] - mean[c] * inv;
    }
    __syncthreads();

    // ---- Phase 1: BN + ReLU + 2x2 avg-pool -> p in LDS --------------------
    for (int i = tid; i < ROWS * CH; i += NTHREAD) {
        const int r = i >> 11;          // / CH
        const int c = i & (CH - 1);
        const size_t off = ((size_t)(b0 + r) * CH + c) * 4;
        const float4 xv = *(const float4*)(x + off);
        const float inv = sh_inv[c];
        const float sh  = sh_shift[c];
        float s = fmaxf(fmaf(xv.x, inv, sh), 0.f)
                + fmaxf(fmaf(xv.y, inv, sh), 0.f)
                + fmaxf(fmaf(xv.z, inv, sh), 0.f)
                + fmaxf(fmaf(xv.w, inv, sh), 0.f);
        p_lds[r * PSTR + c] = 0.25f * s;
    }
    __syncthreads();

    const int wave = tid >> 5;
    const int lane = tid & 31;
    const int half = lane >> 4;   // 0: lanes 0-15, 1: lanes 16-31
    const int l16  = lane & 15;

    // ---- Phase 2: h = relu(p @ w1^T + b1), WMMA f32 16x16x4 ---------------
    // 8 N-tiles of h (N=CRD=128): one per wave. K = CH = 2048.
    {
        const int n0 = wave * 16;
        v8f acc = {};
        for (int k = 0; k < CH; k += 4) {
            const int ka = k + 2 * half;
            // A: M = l16 (rows of p), K = ka..ka+1
            v2f a = *(const v2f*)(p_lds + l16 * PSTR + ka);
            // B: N = n0+l16, K = ka..ka+1  (B[k][n] = w1[n][k])
            v2f b = *(const v2f*)(w1 + (size_t)(n0 + l16) * CH + ka);
            acc = __builtin_amdgcn_wmma_f32_16x16x4_f32(
                false, a, false, b, (short)0, acc, false, false);
        }
        // D layout: lane holds N=l16, M = half*8 + r  (r = vgpr index)
        const float bias = b1[n0 + l16];
#pragma unroll
        for (int r = 0; r < 8; ++r) {
            const int m = half * 8 + r;
            h_lds[m * CRD + n0 + l16] = fmaxf(acc[r] + bias, 0.f);
        }
    }
    __syncthreads();

    // ---- Phase 3: g = sigmoid(h @ w2^T + b2), overwrite p_lds -------------
    // 128 N-tiles over C=2048, 16 tiles per wave. K = CRD = 128.
    for (int t = wave; t < CH / 16; t += NWAVES) {
        const int c0 = t * 16;
        v8f acc = {};
#pragma unroll
        for (int k = 0; k < CRD; k += 4) {
            const int ka = k + 2 * half;
            v2f a = *(const v2f*)(h_lds + l16 * CRD + ka);
            v2f b = *(const v2f*)(w2 + (size_t)(c0 + l16) * CRD + ka);
            acc = __builtin_amdgcn_wmma_f32_16x16x4_f32(
                false, a, false, b, (short)0, acc, false, false);
        }
        const float bias = b2[c0 + l16];
#pragma unroll
        for (int r = 0; r < 8; ++r) {
            const int m = half * 8 + r;
            const float v = acc[r] + bias;
            p_lds[m * PSTR + c0 + l16] = 1.0f / (1.0f + __expf(-v));
        }
    }
    __syncthreads();

    // ---- Phase 4: out = x + g (broadcast over 2x2 spatial) ----------------
    for (int i = tid; i < ROWS * CH; i += NTHREAD) {
        const int r = i >> 11;
        const int c = i & (CH - 1);
        const size_t off = ((size_t)(b0 + r) * CH + c) * 4;
        const float4 xv = *(const float4*)(x + off);   // L2 re-hit
        const float g = p_lds[r * PSTR + c];
        float4 ov;
        ov.x = xv.x + g; ov.y = xv.y + g; ov.z = xv.z + g; ov.w = xv.w + g;
        *(float4*)(out + off) = ov;
    }
}

extern "C" void kernel_launch(void* const* d_in, const int* in_sizes, int n_in,
                              void* d_out, int out_size, void* d_ws, size_t ws_size,
                              hipStream_t stream) {
    (void)in_sizes; (void)n_in; (void)out_size; (void)d_ws; (void)ws_size;
    const float* x     = (const float*)d_in[0];
    const float* gamma = (const float*)d_in[1];
    const float* beta  = (const float*)d_in[2];
    const float* mean  = (const float*)d_in[3];
    const float* var   = (const float*)d_in[4];
    const float* w1    = (const float*)d_in[5];
    const float* b1    = (const float*)d_in[6];
    const float* w2    = (const float*)d_in[7];
    const float* b2    = (const float*)d_in[8];
    float* out = (float*)d_out;

    const dim3 grid(BATCH / ROWS);           // 512 blocks
    const dim3 block(NTHREAD);               // 8 wave32s
    const size_t smem_bytes = SMEM_FLOATS * sizeof(float);  // ~152 KB (<320 KB WGP LDS)
    se_fused_kernel<<<grid, block, smem_bytes, stream>>>(
        x, gamma, beta, mean, var, w1, b1, w2, b2, out);
}